// GridEncoder_16956530885039
// MI455X (gfx1250) — compile-verified
//
#include <hip/hip_runtime.h>
#include <stdint.h>

// ---------------------------------------------------------------------------
// InstantNGP-style multiresolution hash grid encoder for gfx1250 (MI455X).
//   B = 262144 points, 3D input, 16 levels, 2 features/level -> [B, 32] f32.
// All levels use the hash path (side^3 > hashmap_size for every level) and
// every hashmap size is a power of two, so modulo becomes a mask.
// Level 0's 32 KB table is async-DMA'd into LDS (global_load_async_to_lds_b128
// + s_wait_asynccnt) and gathered with ds_load_b64.
// ---------------------------------------------------------------------------

#define NLEV 16

// OFFSETS[l] (entry index into embeddings, each entry = 2 floats)
__device__ constexpr uint32_t kOff[NLEV] = {
    0u,       4096u,    36864u,   299008u,
    823296u,  1347584u, 1871872u, 2396160u,
    2920448u, 3444736u, 3969024u, 4493312u,
    5017600u, 5541888u, 6066176u, 6590464u};

// hashmap_size - 1 (all power-of-two sizes: 4096, 32768, 262144, 524288x13)
__device__ constexpr uint32_t kMask[NLEV] = {
    4095u,   32767u,  262143u, 524287u,
    524287u, 524287u, 524287u, 524287u,
    524287u, 524287u, 524287u, 524287u,
    524287u, 524287u, 524287u, 524287u};

#define PRIME_Y 2654435761u
#define PRIME_Z 805459861u

__global__ __launch_bounds__(256) void grid_encode_kernel(
    const float* __restrict__ xin,   // [B, 3]
    const float* __restrict__ emb,   // [7114752, 2]
    float* __restrict__ out,         // [B, 32]
    int B)
{
    // ---- Stage 0: async-DMA level-0 table (4096 x float2 = 32 KB) into LDS.
    __shared__ float2 lds0[4096];
    {
        const uint32_t t        = threadIdx.x;
        const uint32_t lds_base = (uint32_t)(size_t)(&lds0[0]);
        const uint64_t gbase    = (uint64_t)(size_t)emb;  // level 0 is at offset 0
        // 256 threads x 16 B x 8 passes = 32 KB
#pragma unroll
        for (int p = 0; p < 8; ++p) {
            uint32_t byteoff = (uint32_t)(p * 256 + (int)t) * 16u;
            uint32_t ldsaddr = lds_base + byteoff;
            uint64_t gaddr   = gbase + byteoff;
            asm volatile("global_load_async_to_lds_b128 %0, %1, off"
                         :
                         : "v"(ldsaddr), "v"(gaddr)
                         : "memory");
        }
        asm volatile("s_wait_asynccnt 0" ::: "memory");
    }
    __syncthreads();

    const int i = (int)(blockIdx.x * 256u + threadIdx.x);
    if (i >= B) return;

    const float x = xin[3 * i + 0];
    const float y = xin[3 * i + 1];
    const float z = xin[3 * i + 2];

    float acc[2 * NLEV];

#pragma unroll
    for (int l = 0; l < NLEV; ++l) {
        const float    scale = (float)(16u << l) - 1.0f;
        const uint32_t mask  = kMask[l];

        // Match reference: separate mul then add (no FMA contraction) so the
        // floor / frac split is bit-identical to inputs*scale + 0.5 in f32.
        const float px = __fmul_rn(x, scale) + 0.5f;
        const float py = __fmul_rn(y, scale) + 0.5f;
        const float pz = __fmul_rn(z, scale) + 0.5f;

        const float flx = floorf(px), fly = floorf(py), flz = floorf(pz);
        const float fx = px - flx, fy = py - fly, fz = pz - flz;

        const uint32_t gx = (uint32_t)flx;
        const uint32_t gy = (uint32_t)fly;
        const uint32_t gz = (uint32_t)flz;

        // hash terms; prime for x is 1
        const uint32_t hx0 = gx;             const uint32_t hx1 = gx + 1u;
        const uint32_t hy0 = gy * PRIME_Y;   const uint32_t hy1 = hy0 + PRIME_Y;
        const uint32_t hz0 = gz * PRIME_Z;   const uint32_t hz1 = hz0 + PRIME_Z;

        uint32_t idx[8];
        idx[0] = (hx0 ^ hy0 ^ hz0) & mask;
        idx[1] = (hx1 ^ hy0 ^ hz0) & mask;
        idx[2] = (hx0 ^ hy1 ^ hz0) & mask;
        idx[3] = (hx1 ^ hy1 ^ hz0) & mask;
        idx[4] = (hx0 ^ hy0 ^ hz1) & mask;
        idx[5] = (hx1 ^ hy0 ^ hz1) & mask;
        idx[6] = (hx0 ^ hy1 ^ hz1) & mask;
        idx[7] = (hx1 ^ hy1 ^ hz1) & mask;

        // Issue all 8 corner gathers (b64 each) before consuming.
        float2 v[8];
        if (l == 0) {
#pragma unroll
            for (int c = 0; c < 8; ++c) v[c] = lds0[idx[c]];   // ds_load_b64
        } else {
            const float2* __restrict__ tbl = (const float2*)emb + kOff[l];
#pragma unroll
            for (int c = 0; c < 8; ++c) v[c] = tbl[idx[c]];    // global_load_b64
        }

        const float wx[2] = {1.0f - fx, fx};
        const float wy[2] = {1.0f - fy, fy};
        const float wz[2] = {1.0f - fz, fz};

        float a0 = 0.0f, a1 = 0.0f;
#pragma unroll
        for (int c = 0; c < 8; ++c) {
            const float w = wx[c & 1] * wy[(c >> 1) & 1] * wz[(c >> 2) & 1];
            a0 = fmaf(w, v[c].x, a0);
            a1 = fmaf(w, v[c].y, a1);
        }
        acc[2 * l + 0] = a0;
        acc[2 * l + 1] = a1;
    }

    // 8 x 16-byte stores per point (row is 128 B, 128 B aligned).
    float4* o4 = (float4*)(out + (size_t)i * (2 * NLEV));
#pragma unroll
    for (int k = 0; k < 8; ++k) {
        o4[k] = make_float4(acc[4 * k + 0], acc[4 * k + 1],
                            acc[4 * k + 2], acc[4 * k + 3]);
    }
}

extern "C" void kernel_launch(void* const* d_in, const int* in_sizes, int n_in,
                              void* d_out, int out_size, void* d_ws, size_t ws_size,
                              hipStream_t stream) {
    const float* xin = (const float*)d_in[0];   // [B, 3] f32
    const float* emb = (const float*)d_in[1];   // [7114752, 2] f32
    float*       out = (float*)d_out;           // [B, 32] f32

    const int B = in_sizes[0] / 3;
    const int blocks = (B + 255) / 256;
    hipLaunchKernelGGL(grid_encode_kernel, dim3(blocks), dim3(256), 0, stream,
                       xin, emb, out, B);
}